// Conditional_29222957482793
// MI455X (gfx1250) — compile-verified
//
#include <hip/hip_runtime.h>
#include <hip/hip_bf16.h>

#define N_BOX 4096
#define DDIM  512
#define CCLS  100
#define PPROT 64
#define KCH   128            // K-chunk staged in LDS
#define NCH   (DDIM / KCH)   // 4 chunks
#define LSTR  (KCH + 4)      // padded LDS row stride (132 floats): bank-conflict free
#define MAXT  (CCLS + N_BOX / 16)  // upper bound on #(class,tile) work items = 356

typedef float v2f __attribute__((ext_vector_type(2)));
typedef float v8f __attribute__((ext_vector_type(8)));

// ---------------- prep kernels: bucket boxes by class ----------------

__global__ void k_zero(int* __restrict__ cnt, int* __restrict__ cur) {
  int t = threadIdx.x;
  if (t < CCLS) { cnt[t] = 0; cur[t] = 0; }
}

__global__ void k_count(const int* __restrict__ cls, int* __restrict__ cnt) {
  int n = blockIdx.x * blockDim.x + threadIdx.x;
  if (n < N_BOX) atomicAdd(&cnt[cls[n]], 1);
}

__global__ void k_scan(const int* __restrict__ cnt, int* __restrict__ offs,
                       int* __restrict__ ntile, int* __restrict__ tiles) {
  if (threadIdx.x == 0 && blockIdx.x == 0) {
    int s = 0, k = 0;
    for (int c = 0; c < CCLS; ++c) {
      offs[c] = s;
      int m = cnt[c];
      for (int t = 0; t * 16 < m; ++t) tiles[k++] = (c << 16) | t;
      s += m;
    }
    *ntile = k;
  }
}

__global__ void k_scatter(const int* __restrict__ cls, const int* __restrict__ offs,
                          int* __restrict__ cur, int* __restrict__ bucket) {
  int n = blockIdx.x * blockDim.x + threadIdx.x;
  if (n < N_BOX) {
    int c = cls[n];
    int pos = offs[c] + atomicAdd(&cur[c], 1);
    bucket[pos] = n;  // per-box results do not depend on position -> deterministic
  }
}

// ---------------- main kernel ----------------

__device__ __forceinline__ void proto_scores(const float* __restrict__ sDot,
                                             const float* __restrict__ sL1,
                                             const float* __restrict__ sPn,
                                             int row, int p, float f2, float fn,
                                             float& s0, float& s1, float& s2) {
  float dot = sDot[row * PPROT + p];
  float p2  = sPn[p];
  float pn  = fmaxf(sqrtf(p2), 1e-8f);
  float d0  = 1.0f - dot / (fn * pn);                  // cosine distance
  float d1  = sL1[row * PPROT + p] * (1.0f / DDIM);    // L1 mean
  float d2  = (f2 - 2.0f * dot + p2) * (1.0f / DDIM);  // L2 mean via norms
  s0 = 1.0f / (d0 + 1e-5f);
  s1 = 1.0f / (d1 + 1e-5f);
  s2 = 1.0f / (d2 + 1e-5f);
}

__global__ __launch_bounds__(128) void k_main(
    const float* __restrict__ feats, const float* __restrict__ protos,
    const int* __restrict__ plab,
    const int* __restrict__ cnt, const int* __restrict__ offs,
    const int* __restrict__ ntile, const int* __restrict__ tiles,
    const int* __restrict__ bucket, float* __restrict__ out)
{
  __shared__ float sF[16 * LSTR];       // 16 box rows  x 128-K chunk
  __shared__ float sP[PPROT * LSTR];    // 64 protos    x 128-K chunk
  __shared__ float sDot[16 * PPROT];
  __shared__ float sL1[16 * PPROT];
  __shared__ float sPartP[128];
  __shared__ float sPartF[32];
  __shared__ float sPn[PPROT];
  __shared__ float sFn[16];
  __shared__ int   rowN[16];

  const int nt = *ntile;
  if ((int)blockIdx.x >= nt) return;

  const int info   = tiles[blockIdx.x];
  const int c      = info >> 16;
  const int t      = info & 0xffff;
  const int count  = cnt[c];
  const int base   = offs[c] + t * 16;
  const int nvalid = min(16, count - t * 16);

  const int tid  = threadIdx.x;
  const int lane = tid & 31;
  const int w    = tid >> 5;   // wave id 0..3: owns proto columns [w*16, w*16+16)
  const int idx  = lane & 15;
  const int hi   = lane >> 4;

  if (tid < 16) rowN[tid] = bucket[base + (tid < nvalid ? tid : 0)];
  __syncthreads();

  v8f acc0 = {0.f, 0.f, 0.f, 0.f, 0.f, 0.f, 0.f, 0.f};
  v8f acc1 = {0.f, 0.f, 0.f, 0.f, 0.f, 0.f, 0.f, 0.f};
  float la[8];
#pragma unroll
  for (int j = 0; j < 8; ++j) la[j] = 0.f;
  float pnAcc = 0.f, fnAcc = 0.f;

  const float* pBase = protos + (size_t)c * (PPROT * DDIM);
  const int cb = w * 16;
  const int cj = cb + hi * 8;

  for (int ck = 0; ck < NCH; ++ck) {
    const int kb0 = ck * KCH;
    // stage 16x128 feat chunk (float4, coalesced)
    for (int q = tid; q < 16 * (KCH / 4); q += 128) {
      int r = q >> 5, x = q & 31;
      float4 v = *(const float4*)(feats + (size_t)rowN[r] * DDIM + kb0 + x * 4);
      *(float4*)&sF[r * LSTR + x * 4] = v;
    }
    // stage 64x128 proto chunk (float4, coalesced; L2-resident)
    for (int q = tid; q < PPROT * (KCH / 4); q += 128) {
      int r = q >> 5, x = q & 31;
      float4 v = *(const float4*)(pBase + (size_t)r * DDIM + kb0 + x * 4);
      *(float4*)&sP[r * LSTR + x * 4] = v;
    }
    // prefetch next chunk while this one is consumed (global_prefetch_b8)
    if (ck + 1 < NCH) {
      const int kb1 = kb0 + KCH;
      if (tid < PPROT) __builtin_prefetch(pBase + (size_t)tid * DDIM + kb1, 0, 0);
      if (tid >= PPROT && tid < PPROT + 16)
        __builtin_prefetch(feats + (size_t)rowN[tid - PPROT] * DDIM + kb1, 0, 0);
    }
    __syncthreads();

    // ---- dot products via f32 WMMA: 32 k-steps of 16x16x4 per chunk ----
    // two accumulators break the WMMA->WMMA RAW chain on C/D
#pragma unroll
    for (int kk = 0; kk < KCH / 4; kk += 2) {
      {
        const int kb = kk * 4 + 2 * hi;
        v2f a, b;
        a.x = sF[idx * LSTR + kb];
        a.y = sF[idx * LSTR + kb + 1];
        b.x = sP[(cb + idx) * LSTR + kb];
        b.y = sP[(cb + idx) * LSTR + kb + 1];
        acc0 = __builtin_amdgcn_wmma_f32_16x16x4_f32(
            false, a, false, b, (short)0, acc0, false, false);
      }
      {
        const int kb = (kk + 1) * 4 + 2 * hi;
        v2f a, b;
        a.x = sF[idx * LSTR + kb];
        a.y = sF[idx * LSTR + kb + 1];
        b.x = sP[(cb + idx) * LSTR + kb];
        b.y = sP[(cb + idx) * LSTR + kb + 1];
        acc1 = __builtin_amdgcn_wmma_f32_16x16x4_f32(
            false, a, false, b, (short)0, acc1, false, false);
      }
    }

    // ---- L1: lane owns (row=idx, 8 cols); float4 LDS reads, broadcast-friendly ----
    for (int d4 = 0; d4 < KCH / 4; ++d4) {
      float4 f4 = *(const float4*)&sF[idx * LSTR + d4 * 4];
#pragma unroll
      for (int j = 0; j < 8; ++j) {
        float4 p4 = *(const float4*)&sP[(cj + j) * LSTR + d4 * 4];
        la[j] += fabsf(f4.x - p4.x) + fabsf(f4.y - p4.y) +
                 fabsf(f4.z - p4.z) + fabsf(f4.w - p4.w);
      }
    }

    // ---- norm partials ----
    {
      const int p = tid & 63, h = tid >> 6;
      const float* pp = &sP[p * LSTR + h * 64];
#pragma unroll 8
      for (int d = 0; d < 64; ++d) { float v = pp[d]; pnAcc += v * v; }
    }
    if (tid < 32) {
      const int r = tid & 15, h = tid >> 4;
      const float* fp = &sF[r * LSTR + h * 64];
#pragma unroll 8
      for (int d = 0; d < 64; ++d) { float v = fp[d]; fnAcc += v * v; }
    }
    __syncthreads();
  }

  // C/D fragment: element (M = r + 8*hi, N = idx) in vgpr r
#pragma unroll
  for (int r = 0; r < 8; ++r)
    sDot[(r + 8 * hi) * PPROT + cb + idx] = acc0[r] + acc1[r];
#pragma unroll
  for (int j = 0; j < 8; ++j)
    sL1[idx * PPROT + cj + j] = la[j];
  sPartP[tid] = pnAcc;
  if (tid < 32) sPartF[tid] = fnAcc;
  __syncthreads();
  if (tid < PPROT) sPn[tid] = sPartP[tid] + sPartP[tid + 64];
  if (tid < 16)    sFn[tid] = sPartF[tid] + sPartF[tid + 16];
  __syncthreads();

  // ---- epilogue: per-box 3x softmax over P, mean, argmax, label ----
  if (tid < nvalid) {
    const int n  = rowN[tid];
    const float f2 = sFn[tid];
    const float fn = fmaxf(sqrtf(f2), 1e-8f);

    float mx0 = -3.4e38f, mx1 = -3.4e38f, mx2 = -3.4e38f;
    for (int p = 0; p < PPROT; ++p) {
      float s0, s1, s2;
      proto_scores(sDot, sL1, sPn, tid, p, f2, fn, s0, s1, s2);
      mx0 = fmaxf(mx0, s0); mx1 = fmaxf(mx1, s1); mx2 = fmaxf(mx2, s2);
    }
    float sm0 = 0.f, sm1 = 0.f, sm2 = 0.f;
    for (int p = 0; p < PPROT; ++p) {
      float s0, s1, s2;
      proto_scores(sDot, sL1, sPn, tid, p, f2, fn, s0, s1, s2);
      sm0 += expf(s0 - mx0); sm1 += expf(s1 - mx1); sm2 += expf(s2 - mx2);
    }
    const float r0 = 1.f / sm0, r1 = 1.f / sm1, r2 = 1.f / sm2;
    float best = -1.0f; int bi = 0;
    for (int p = 0; p < PPROT; ++p) {
      float s0, s1, s2;
      proto_scores(sDot, sL1, sPn, tid, p, f2, fn, s0, s1, s2);
      float pr = (expf(s0 - mx0) * r0 + expf(s1 - mx1) * r1 + expf(s2 - mx2) * r2)
                 * (1.0f / 3.0f);
      out[N_BOX + (size_t)n * PPROT + p] = pr;
      if (pr > best) { best = pr; bi = p; }   // strict > keeps first max (jnp.argmax)
    }
    out[n] = (float)plab[c * PPROT + bi];
  }
}

// ---------------- launcher ----------------

extern "C" void kernel_launch(void* const* d_in, const int* in_sizes, int n_in,
                              void* d_out, int out_size, void* d_ws, size_t ws_size,
                              hipStream_t stream) {
  (void)in_sizes; (void)n_in; (void)out_size; (void)ws_size;
  const float* feats  = (const float*)d_in[0];
  const float* protos = (const float*)d_in[1];
  const int*   cls    = (const int*)d_in[2];
  const int*   plab   = (const int*)d_in[3];
  float* out = (float*)d_out;

  int* ws     = (int*)d_ws;
  int* cnt    = ws;         // [0,100)
  int* offs   = ws + 128;   // [128,228)
  int* cur    = ws + 256;   // [256,356)
  int* ntile  = ws + 384;   // [384]
  int* tiles  = ws + 400;   // [400,756)
  int* bucket = ws + 768;   // [768,4864)  ~19.5 KB total

  k_zero   <<<1, 128, 0, stream>>>(cnt, cur);
  k_count  <<<N_BOX / 256, 256, 0, stream>>>(cls, cnt);
  k_scan   <<<1, 1, 0, stream>>>(cnt, offs, ntile, tiles);
  k_scatter<<<N_BOX / 256, 256, 0, stream>>>(cls, offs, cur, bucket);
  k_main   <<<MAXT, 128, 0, stream>>>(feats, protos, plab, cnt, offs, ntile, tiles,
                                      bucket, out);
}